// ChamferPccRateDistortionLoss_44418551775684
// MI455X (gfx1250) — compile-verified
//
#include <hip/hip_runtime.h>

// CDNA5 / gfx1250, wave32.
typedef __attribute__((ext_vector_type(2))) float v2f;
typedef __attribute__((ext_vector_type(8))) float v8f;

#define BB 4
#define NN 8192

// Pack (p0,p1,p2) -> (p0,p1,p2,|p|^2) for fast WMMA operand loads.
__global__ __launch_bounds__(256) void pack4_kernel(const float* __restrict__ src,
                                                    float4* __restrict__ dst, int n) {
  int i = blockIdx.x * blockDim.x + threadIdx.x;
  if (i < n) {
    const float* s = src + (size_t)i * 3;
    float a = s[0], b = s[1], c = s[2];
    float4 o;
    o.x = a; o.y = b; o.z = c; o.w = a * a + b * b + c * c;
    dst[i] = o;
  }
}

// Each wave owns 16 rows of P for one batch; sweeps all columns of Q in 16-wide
// tiles. One V_WMMA_F32_16X16X4_F32 produces the full 16x16 squared-distance
// tile:  D = A x B + C  with
//   A row i = (-2 p_i0, -2 p_i1, -2 p_i2, 1)       (K=4; C=3 padded)
//   B col j = ( q_j0,    q_j1,    q_j2,  |q_j|^2)
//   C[i][j] = |p_i|^2
// Row-min tracked in 8 VGPRs, reduced across the 16-lane half, stored directly.
__global__ __launch_bounds__(256) void chamfer_rowmin_wmma(
    const float4* __restrict__ P4, const float4* __restrict__ Q4,
    float* __restrict__ dmin) {
  const int lane = threadIdx.x & 31;
  const int wid  = (blockIdx.x * blockDim.x + threadIdx.x) >> 5;
  const int tiles = NN / 16;            // 512 column tiles
  const int b   = wid / tiles;
  const int r0  = (wid % tiles) * 16;   // this wave's row tile
  const int hi  = lane >> 4;            // 0: K=0,1 half   1: K=2,3 half
  const int l15 = lane & 15;

  const float4* Pb = P4 + (size_t)b * NN;
  const float4* Qb = Q4 + (size_t)b * NN;

  // A operand (constant across the sweep). f32 16x4 A layout:
  // lanes 0-15 hold (K=0,K=1) for M=lane; lanes 16-31 hold (K=2,K=3).
  float4 p = Pb[r0 + l15];
  v2f A;
  A.x = hi ? (-2.0f * p.z) : (-2.0f * p.x);
  A.y = hi ? 1.0f          : (-2.0f * p.y);

  // C operand: C[r] = |p_{r0 + r + 8*hi}|^2  (16x16 f32 C/D layout).
  v8f Cm;
#pragma unroll
  for (int r = 0; r < 8; ++r) Cm[r] = Pb[r0 + r + 8 * hi].w;

  v8f m;
#pragma unroll
  for (int r = 0; r < 8; ++r) m[r] = __builtin_inff();

  // B operand pointer: lane l15 reads column (jt*16 + l15); lanes 0-15 take
  // float2 components (q0,q1), lanes 16-31 take (q2,|q|^2).
  const v2f* qptr = (const v2f*)(Qb + l15) + hi;

#pragma unroll 4
  for (int jt = 0; jt < tiles; ++jt) {
    v2f Bm = *qptr;
    qptr += 32;  // 16 float4 per tile = 32 v2f
    v8f D = __builtin_amdgcn_wmma_f32_16x16x4_f32(
        /*neg_a=*/false, A, /*neg_b=*/false, Bm,
        /*c_mod=*/(short)0, Cm, /*reuse_a=*/false, /*reuse_b=*/false);
#pragma unroll
    for (int r = 0; r < 8; ++r) m[r] = __builtin_fminf(m[r], D[r]);
  }

  // Min across the 16 lanes of each half (rows r+8*hi live per-half).
#pragma unroll
  for (int mask = 1; mask <= 8; mask <<= 1) {
#pragma unroll
    for (int r = 0; r < 8; ++r) {
      float o = __shfl_xor(m[r], mask, 32);
      m[r] = __builtin_fminf(m[r], o);
    }
  }

  if (l15 == 0) {
    float* out = dmin + (size_t)b * NN + r0 + 8 * hi;
#pragma unroll
    for (int r = 0; r < 8; ++r) out[r] = m[r];
  }
}

// loss = mean(dist1) + mean(dist2) = (sum over 2*B*N mins) / (B*N)
__global__ __launch_bounds__(1024) void reduce_mean_kernel(const float* __restrict__ d,
                                                           float* __restrict__ out) {
  const int n = 2 * BB * NN;  // 65536
  float s = 0.0f;
  for (int i = threadIdx.x; i < n; i += 1024) s += d[i];

  __shared__ float sm[32];
#pragma unroll
  for (int mask = 16; mask >= 1; mask >>= 1) s += __shfl_xor(s, mask, 32);
  int lane = threadIdx.x & 31, w = threadIdx.x >> 5;
  if (lane == 0) sm[w] = s;
  __syncthreads();
  if (w == 0) {
    s = sm[lane];
#pragma unroll
    for (int mask = 16; mask >= 1; mask >>= 1) s += __shfl_xor(s, mask, 32);
    if (lane == 0) out[0] = s * (1.0f / (float)(BB * NN));
  }
}

extern "C" void kernel_launch(void* const* d_in, const int* in_sizes, int n_in,
                              void* d_out, int out_size, void* d_ws, size_t ws_size,
                              hipStream_t stream) {
  (void)in_sizes; (void)n_in; (void)out_size; (void)ws_size;
  const float* target = (const float*)d_in[0];  // [B,N,3] f32
  const float* xhat   = (const float*)d_in[1];  // [B,N,3] f32

  char* ws = (char*)d_ws;
  const int n = BB * NN;                               // 32768 points per array
  float4* X4  = (float4*)ws;                           // 512 KB
  float4* Y4  = (float4*)(ws + (size_t)n * 16);        // 512 KB
  float* dist = (float*)(ws + (size_t)2 * n * 16);     // dist1 | dist2 (256 KB)

  pack4_kernel<<<(n + 255) / 256, 256, 0, stream>>>(target, X4, n);
  pack4_kernel<<<(n + 255) / 256, 256, 0, stream>>>(xhat,   Y4, n);

  const int totalWaves = BB * (NN / 16);  // 2048 row tiles
  dim3 grid(totalWaves / 8), block(256);  // 8 waves per block
  chamfer_rowmin_wmma<<<grid, block, 0, stream>>>(X4, Y4, dist);      // dist1
  chamfer_rowmin_wmma<<<grid, block, 0, stream>>>(Y4, X4, dist + n);  // dist2

  reduce_mean_kernel<<<1, 1024, 0, stream>>>(dist, (float*)d_out);
}